// LightGCN_42451456754103
// MI455X (gfx1250) — compile-verified
//
#include <hip/hip_runtime.h>

// ---------------------------------------------------------------------------
// LightGCN-style forward on MI455X (gfx1250).
//
// 4x SpMM (scatter-add) + 3 embedding copies. Memory/atomic bound
// (0.25 FLOP/byte) -> WMMA cannot be fed (random indices, no dense tiles).
// CDNA5 features used instead:
//   * global_load_async_to_lds_b32 (ASYNCcnt) double-buffered edge streaming
//   * native global_atomic_add_f32 scatter into L2-resident segments
//   * wave32-native layout: 2 edges/wave, 16 lanes x float4 = 64-dim row
//   * non-temporal stores for the passthrough copies so the 53.8MB of dead
//     output data does not evict the gather-hot tables from the 192MB L2
// ---------------------------------------------------------------------------

#define WAVES_PER_BLOCK 8
#define SPMM_BLOCKS 1024
#define CHUNK 32  // edges staged per async group (one lane per edge triple)

// clang-native 128-bit vector (accepted by __builtin_nontemporal_store).
typedef float floatx4 __attribute__((ext_vector_type(4)));

// Low 32 bits of a generic pointer into LDS == LDS byte address (aperture
// layout: addr[31:0] is the LDS offset).
__device__ __forceinline__ unsigned lds_byte_off(const void* p) {
  return (unsigned)(unsigned long long)p;
}

// CDNA5 async copy: each enabled lane moves 4B global -> LDS, tracked by
// ASYNCcnt (per-wave). VDST = LDS byte address VGPR, VADDR = 64-bit global.
__device__ __forceinline__ void async_ld_b32(unsigned ldsByte, const void* g) {
  asm volatile("global_load_async_to_lds_b32 %0, %1, off"
               :: "v"(ldsByte), "v"(g)
               : "memory");
}

__device__ __forceinline__ void atomic_add_f32(float* p, float v) {
  __hip_atomic_fetch_add(p, v, __ATOMIC_RELAXED, __HIP_MEMORY_SCOPE_AGENT);
}

// out[rows[e]*64 + d] += vals[e] * X[cols[e]*64 + d]   for d in [0,64)
__global__ void __launch_bounds__(WAVES_PER_BLOCK * 32)
spmm_atomic_kernel(const int* __restrict__ rows,
                   const int* __restrict__ cols,
                   const float* __restrict__ vals,
                   const float* __restrict__ X,
                   float* __restrict__ out,
                   int nnz) {
  // [wave][buffer][rows|cols|vals][lane]
  __shared__ unsigned sh[WAVES_PER_BLOCK][2][3][CHUNK];

  const int lane   = threadIdx.x & 31;
  const int half   = lane >> 4;   // which edge of the pair this lane serves
  const int l16    = lane & 15;   // 16 lanes x float4 = 64 dims
  const int wave   = threadIdx.x >> 5;
  const int gwave  = blockIdx.x * WAVES_PER_BLOCK + wave;
  const int nwaves = gridDim.x * WAVES_PER_BLOCK;
  const int nchunks = nnz >> 5;

  auto stage = [&](int c, int b) {
    const int base = c << 5;  // 32 edges per chunk, lane L stages edge base+L
    async_ld_b32(lds_byte_off(&sh[wave][b][0][lane]), rows + base + lane);
    async_ld_b32(lds_byte_off(&sh[wave][b][1][lane]), cols + base + lane);
    async_ld_b32(lds_byte_off(&sh[wave][b][2][lane]), vals + base + lane);
  };

  int c = gwave;
  if (c < nchunks) {
    int buf = 0;
    stage(c, buf);  // prime the pipeline
    for (; c < nchunks; c += nwaves) {
      const int cn = c + nwaves;
      if (cn < nchunks) {
        stage(cn, buf ^ 1);  // prefetch next chunk into the other buffer
        // 3 newest (next chunk) may still be in flight; current chunk done.
        asm volatile("s_wait_asynccnt 3" ::: "memory");
      } else {
        asm volatile("s_wait_asynccnt 0" ::: "memory");
      }

      // 2 edges per iteration: lanes 0-15 -> edge 2i, lanes 16-31 -> 2i+1.
      // LDS reads are a 2-address broadcast (adjacent words, conflict-free).
      #pragma unroll 4
      for (int i = 0; i < (CHUNK >> 1); ++i) {
        const int e = (i << 1) + half;
        const int   r = (int)sh[wave][buf][0][e];
        const int   k = (int)sh[wave][buf][1][e];
        const float v = __uint_as_float(sh[wave][buf][2][e]);

        // 512B/wave fully-coalesced gather covering both edges' rows.
        const floatx4 x = *(const floatx4*)(X + ((size_t)k << 6) + (l16 << 2));
        float* o = out + ((size_t)r << 6) + (l16 << 2);
        atomic_add_f32(o + 0, v * x.x);
        atomic_add_f32(o + 1, v * x.y);
        atomic_add_f32(o + 2, v * x.z);
        atomic_add_f32(o + 3, v * x.w);
      }
      buf ^= 1;
    }
  }

  // Tail (nnz % 32) — empty for NNZ=1e6 but kept for generality.
  if (gwave == 0) {
    for (int e = (nchunks << 5); e < nnz; ++e) {
      const int   r = rows[e];
      const int   k = cols[e];
      const float v = vals[e];
      const float2 x = *(const float2*)(X + ((size_t)k << 6) + (lane << 1));
      float* o = out + ((size_t)r << 6) + (lane << 1);
      atomic_add_f32(o + 0, v * x.x);
      atomic_add_f32(o + 1, v * x.y);
    }
  }
}

__global__ void zero_f4_kernel(floatx4* __restrict__ dst, long long n4) {
  long long i = (long long)blockIdx.x * blockDim.x + threadIdx.x;
  const long long stride = (long long)gridDim.x * blockDim.x;
  const floatx4 z = {0.f, 0.f, 0.f, 0.f};
  for (; i < n4; i += stride) dst[i] = z;
}

// Loads stay RT (they pre-warm L2 with rows the SpMM gathers will reuse);
// stores are non-temporal (copied data is never re-read -> don't pollute L2).
__global__ void copy_f4_nt_kernel(const floatx4* __restrict__ src,
                                  floatx4* __restrict__ dst, long long n4) {
  long long i = (long long)blockIdx.x * blockDim.x + threadIdx.x;
  const long long stride = (long long)gridDim.x * blockDim.x;
  for (; i < n4; i += stride) {
    const floatx4 v = src[i];
    __builtin_nontemporal_store(v, dst + i);
  }
}

extern "C" void kernel_launch(void* const* d_in, const int* in_sizes, int n_in,
                              void* d_out, int out_size, void* d_ws, size_t ws_size,
                              hipStream_t stream) {
  const int*   uj_rows = (const int*)d_in[0];
  const int*   uj_cols = (const int*)d_in[1];
  const float* uj_vals = (const float*)d_in[2];
  const int*   ij_rows = (const int*)d_in[3];
  const int*   ij_cols = (const int*)d_in[4];
  const float* ij_vals = (const float*)d_in[5];
  const float* user_emb   = (const float*)d_in[6];
  const float* top_emb    = (const float*)d_in[7];
  const float* bottom_emb = (const float*)d_in[8];

  const int nnz_uj = in_sizes[0];
  const int nnz_ij = in_sizes[3];
  const long long U = in_sizes[6] / 64;  // 100000
  const long long T = in_sizes[7] / 64;  // 50000
  const long long B = in_sizes[8] / 64;  // 60000

  float* out = (float*)d_out;
  float* user_temp = out;                       // [U,64]
  float* top_temp  = user_temp + U * 64;        // [T,64]
  float* bot_temp  = top_temp + T * 64;         // [B,64]
  float* top_copy  = bot_temp + B * 64;         // [T,64]
  float* bot_copy  = top_copy + T * 64;         // [B,64]
  float* user_copy = bot_copy + B * 64;         // [U,64]

  const dim3 blk(256);

  // 1) Zero the three accumulated segments (contiguous prefix).
  {
    const long long n4 = (U + T + B) * 16;  // 64 floats/row = 16 float4
    zero_f4_kernel<<<2048, blk, 0, stream>>>((floatx4*)user_temp, n4);
  }

  // 2) Passthrough copies (independent regions, NT stores).
  copy_f4_nt_kernel<<<1024, blk, 0, stream>>>((const floatx4*)top_emb,
                                              (floatx4*)top_copy, T * 16);
  copy_f4_nt_kernel<<<1024, blk, 0, stream>>>((const floatx4*)bottom_emb,
                                              (floatx4*)bot_copy, B * 16);
  copy_f4_nt_kernel<<<1024, blk, 0, stream>>>((const floatx4*)user_emb,
                                              (floatx4*)user_copy, U * 16);

  // 3) Four SpMMs (stream order guarantees zero-init completes first).
  // user_temp = A_UJ @ bottom_emb
  spmm_atomic_kernel<<<SPMM_BLOCKS, blk, 0, stream>>>(
      uj_rows, uj_cols, uj_vals, bottom_emb, user_temp, nnz_uj);
  // bot_temp += A_UJ^T @ user_emb
  spmm_atomic_kernel<<<SPMM_BLOCKS, blk, 0, stream>>>(
      uj_cols, uj_rows, uj_vals, user_emb, bot_temp, nnz_uj);
  // top_temp = A_IJ @ bottom_emb
  spmm_atomic_kernel<<<SPMM_BLOCKS, blk, 0, stream>>>(
      ij_rows, ij_cols, ij_vals, bottom_emb, top_temp, nnz_ij);
  // bot_temp += A_IJ^T @ top_emb
  spmm_atomic_kernel<<<SPMM_BLOCKS, blk, 0, stream>>>(
      ij_cols, ij_rows, ij_vals, top_emb, bot_temp, nnz_ij);
}